// FusedGAT_43654047596707
// MI455X (gfx1250) — compile-verified
//
#include <hip/hip_runtime.h>
#include <math.h>

// ---------------------------------------------------------------------------
// Fused 2-layer GAT forward for MI455X (gfx1250).
// GEMMs use V_WMMA_F32_16X16X4_F32 (exact fp32 matmul on the matrix pipe).
// Edge softmax/aggregation: 3 passes with L2-resident f32/uint atomics.
// ---------------------------------------------------------------------------

typedef float v2f __attribute__((ext_vector_type(2)));
typedef float v8f __attribute__((ext_vector_type(8)));

#define NEG_SLOPE 0.2f
#define GAT_EPS 1e-16f

// Monotonic float <-> uint mapping for atomic max on floats.
__device__ __forceinline__ unsigned fenc(float f) {
  unsigned u = __float_as_uint(f);
  return (u & 0x80000000u) ? ~u : (u | 0x80000000u);
}
__device__ __forceinline__ float fdec(unsigned k) {
  unsigned u = (k & 0x80000000u) ? (k ^ 0x80000000u) : ~k;
  return __uint_as_float(u);
}

// ---------------- GEMM 1: H[N,64] = X[N,128] @ W[128,64] (WMMA f32) --------
// block = 128 threads (4 waves); each block: 16 rows; wave w: cols 16w..16w+15
__global__ void gat_gemm1_wmma(const float* __restrict__ X,
                               const float* __restrict__ W,
                               float* __restrict__ H, int N) {
  const int lane    = threadIdx.x & 31;
  const int wave    = threadIdx.x >> 5;        // 0..3
  const int row0    = blockIdx.x * 16;
  const int colBase = wave * 16;
  const int m       = lane & 15;
  const int khalf   = (lane >> 4) * 2;         // 0 or 2

  int rowA = row0 + m; if (rowA >= N) rowA = N - 1;   // clamp (uniform EXEC)
  const float* xrow = X + (long)rowA * 128;

  v8f c = {};
  for (int k0 = 0; k0 < 128; k0 += 4) {
    const int k = k0 + khalf;
    v2f a; a.x = xrow[k];                 a.y = xrow[k + 1];
    v2f b; b.x = W[k * 64 + colBase + m]; b.y = W[(k + 1) * 64 + colBase + m];
    c = __builtin_amdgcn_wmma_f32_16x16x4_f32(false, a, false, b,
                                              (short)0, c, false, false);
  }
  const int mrow = (lane >> 4) * 8;  // VGPR v: lanes0-15 -> M=v, lanes16-31 -> M=v+8
#pragma unroll
  for (int v = 0; v < 8; ++v) {
    int r = row0 + mrow + v;
    if (r < N) H[(long)r * 64 + colBase + m] = c[v];
  }
}

// ---------------- GEMM 2: H[N,40] = X[N,64] @ W[64,40] (WMMA f32) ----------
// block = 96 threads (3 waves); wave w: cols 16w..16w+15 (cols>=40 masked)
__global__ void gat_gemm2_wmma(const float* __restrict__ X,
                               const float* __restrict__ W,
                               float* __restrict__ H, int N) {
  const int lane    = threadIdx.x & 31;
  const int wave    = threadIdx.x >> 5;        // 0..2
  const int row0    = blockIdx.x * 16;
  const int col     = wave * 16 + (lane & 15);
  const int m       = lane & 15;
  const int khalf   = (lane >> 4) * 2;
  const bool colOK  = (col < 40);

  int rowA = row0 + m; if (rowA >= N) rowA = N - 1;
  const float* xrow = X + (long)rowA * 64;

  v8f c = {};
  for (int k0 = 0; k0 < 64; k0 += 4) {
    const int k = k0 + khalf;
    v2f a; a.x = xrow[k]; a.y = xrow[k + 1];
    v2f b;
    b.x = colOK ? W[k * 40 + col] : 0.0f;
    b.y = colOK ? W[(k + 1) * 40 + col] : 0.0f;
    c = __builtin_amdgcn_wmma_f32_16x16x4_f32(false, a, false, b,
                                              (short)0, c, false, false);
  }
  const int mrow = (lane >> 4) * 8;
#pragma unroll
  for (int v = 0; v < 8; ++v) {
    int r = row0 + mrow + v;
    if (colOK && r < N) H[(long)r * 40 + col] = c[v];
  }
}

// -------- attention coefficients, layer 1: per (node, head) ----------------
__global__ void gat_attn1(const float* __restrict__ H, const float* __restrict__ aS,
                          const float* __restrict__ aD, float* __restrict__ eS,
                          float* __restrict__ eD, int N) {
  int t = blockIdx.x * blockDim.x + threadIdx.x;
  if (t >= N * 8) return;
  const int head = t & 7;
  const float* hp = H + (long)(t >> 3) * 64 + head * 8;
  const float* as = aS + head * 8;
  const float* ad = aD + head * 8;
  float s = 0.f, d = 0.f;
#pragma unroll
  for (int c = 0; c < 8; ++c) { float h = hp[c]; s += h * as[c]; d += h * ad[c]; }
  eS[t] = s; eD[t] = d;
}

// -------- attention coefficients, layer 2: per node ------------------------
__global__ void gat_attn2(const float* __restrict__ H, const float* __restrict__ aS,
                          const float* __restrict__ aD, float* __restrict__ eS,
                          float* __restrict__ eD, int N) {
  int n = blockIdx.x * blockDim.x + threadIdx.x;
  if (n >= N) return;
  const float* hp = H + (long)n * 40;
  float s = 0.f, d = 0.f;
#pragma unroll 8
  for (int c = 0; c < 40; ++c) { float h = hp[c]; s += h * aS[c]; d += h * aD[c]; }
  eS[n] = s; eD[n] = d;
}

// -------- edge pass A: e = leaky_relu(eS[src]+eD[dst]); segment max --------
__global__ void gat_edge_max(const int* __restrict__ src, const int* __restrict__ dst,
                             const float* __restrict__ eS, const float* __restrict__ eD,
                             float* __restrict__ eBuf, unsigned* __restrict__ mKey,
                             int E, int H) {
  int t = blockIdx.x * blockDim.x + threadIdx.x;
  if (t >= E * H) return;
  const int e = t / H, h = t - e * H;
  const int s = src[e], d = dst[e];
  float v = eS[s * H + h] + eD[d * H + h];
  v = (v > 0.f) ? v : v * NEG_SLOPE;
  eBuf[t] = v;
  atomicMax(mKey + d * H + h, fenc(v));
}

// -------- edge pass B: p = exp(e - m[dst]); segment sum --------------------
__global__ void gat_edge_sum(const int* __restrict__ dst, float* __restrict__ eBuf,
                             const unsigned* __restrict__ mKey, float* __restrict__ sSum,
                             int E, int H) {
  int t = blockIdx.x * blockDim.x + threadIdx.x;
  if (t >= E * H) return;
  const int e = t / H, h = t - e * H;
  const int d = dst[e];
  float p = expf(eBuf[t] - fdec(mKey[d * H + h]));
  eBuf[t] = p;
  atomicAdd(sSum + d * H + h, p);
}

// -------- edge pass C, layer 1: scatter h[src]*alpha into agg --------------
__global__ void gat_edge_msg1(const int* __restrict__ src, const int* __restrict__ dst,
                              const float* __restrict__ eBuf, const float* __restrict__ sSum,
                              const float* __restrict__ Hn, float* __restrict__ agg, int E) {
  int t = blockIdx.x * blockDim.x + threadIdx.x;
  if (t >= E * 8) return;
  const int e = t >> 3, h = t & 7;
  const int s = src[e], d = dst[e];
  const float alpha = eBuf[t] / (sSum[d * 8 + h] + GAT_EPS);
  const float4* hp = (const float4*)(Hn + (long)s * 64 + h * 8);
  float4 u0 = hp[0], u1 = hp[1];
  float* o = agg + (long)d * 64 + h * 8;
  atomicAdd(o + 0, u0.x * alpha); atomicAdd(o + 1, u0.y * alpha);
  atomicAdd(o + 2, u0.z * alpha); atomicAdd(o + 3, u0.w * alpha);
  atomicAdd(o + 4, u1.x * alpha); atomicAdd(o + 5, u1.y * alpha);
  atomicAdd(o + 6, u1.z * alpha); atomicAdd(o + 7, u1.w * alpha);
}

// -------- edge pass C, layer 2 (H=1, C=40) ---------------------------------
__global__ void gat_edge_msg2(const int* __restrict__ src, const int* __restrict__ dst,
                              const float* __restrict__ eBuf, const float* __restrict__ sSum,
                              const float* __restrict__ Hn, float* __restrict__ agg, int E) {
  int e = blockIdx.x * blockDim.x + threadIdx.x;
  if (e >= E) return;
  const int s = src[e], d = dst[e];
  const float alpha = eBuf[e] / (sSum[d] + GAT_EPS);
  const float* hp = Hn + (long)s * 40;
  float* o = agg + (long)d * 40;
#pragma unroll 8
  for (int c = 0; c < 40; ++c) atomicAdd(o + c, hp[c] * alpha);
}

// -------- layer-1 finalize: agg = relu(agg + b1) (in place) ----------------
__global__ void gat_bias_relu(float* __restrict__ agg, const float* __restrict__ b, long n64) {
  long t = (long)blockIdx.x * blockDim.x + threadIdx.x;
  if (t >= n64) return;
  float v = agg[t] + b[t & 63];
  agg[t] = v > 0.f ? v : 0.f;
}

// -------- layer-2 finalize: out = log_softmax(agg + b2) --------------------
__global__ void gat_logsoftmax(const float* __restrict__ agg, const float* __restrict__ b,
                               float* __restrict__ out, int N) {
  int n = blockIdx.x * blockDim.x + threadIdx.x;
  if (n >= N) return;
  const float* a = agg + (long)n * 40;
  float v[40];
  float mx = -INFINITY;
#pragma unroll 8
  for (int c = 0; c < 40; ++c) { v[c] = a[c] + b[c]; mx = fmaxf(mx, v[c]); }
  float sum = 0.f;
#pragma unroll 8
  for (int c = 0; c < 40; ++c) sum += expf(v[c] - mx);
  const float ls = mx + logf(sum);
  float* o = out + (long)n * 40;
#pragma unroll 8
  for (int c = 0; c < 40; ++c) o[c] = v[c] - ls;
}

extern "C" void kernel_launch(void* const* d_in, const int* in_sizes, int n_in,
                              void* d_out, int out_size, void* d_ws, size_t ws_size,
                              hipStream_t stream) {
  const float* x     = (const float*)d_in[0];
  const int*   eidx  = (const int*)d_in[1];
  const float* W1    = (const float*)d_in[2];
  const float* aS1   = (const float*)d_in[3];
  const float* aD1   = (const float*)d_in[4];
  const float* b1    = (const float*)d_in[5];
  const float* W2    = (const float*)d_in[6];
  const float* aS2   = (const float*)d_in[7];
  const float* aD2   = (const float*)d_in[8];
  const float* b2    = (const float*)d_in[9];
  float* out = (float*)d_out;

  const int N = in_sizes[0] / 128;
  const int E = in_sizes[1] / 2;
  const int* src = eidx;
  const int* dst = eidx + E;

  const long N64 = (long)N * 64, N40 = (long)N * 40;
  const long N8  = (long)N * 8,  E8  = (long)E * 8;

  // ---- workspace partition (floats) ----
  float* ws      = (float*)d_ws;
  float* h1      = ws;                        // N*64
  float* esrc1   = h1 + N64;                  // N*8
  float* edst1   = esrc1 + N8;                // N*8
  float* e1      = edst1 + N8;                // E*8
  float* s1      = e1 + E8;                   // N*8   (zeroed, contiguous with below)
  unsigned* mK1  = (unsigned*)(s1 + N8);      // N*8
  float* agg1    = (float*)mK1 + N8;          // N*64
  float* h2      = agg1 + N64;                // N*40
  float* esrc2   = h2 + N40;                  // N
  float* edst2   = esrc2 + N;                 // N
  float* e2      = edst2 + N;                 // E
  float* s2      = e2 + E;                    // N     (zeroed, contiguous with below)
  unsigned* mK2  = (unsigned*)(s2 + N);       // N
  float* agg2    = (float*)mK2 + N;           // N*40

  const int BT = 256;
  const int rowBlocks = (N + 15) / 16;

  // ===================== Layer 1 =====================
  gat_gemm1_wmma<<<rowBlocks, 128, 0, stream>>>(x, W1, h1, N);
  gat_attn1<<<(int)((N8 + BT - 1) / BT), BT, 0, stream>>>(h1, aS1, aD1, esrc1, edst1, N);
  // zero s1 | mK1 | agg1 in one contiguous memset (mK1==0 decodes to NaN -> empty-seg m=0)
  hipMemsetAsync(s1, 0, (size_t)(N8 + N8 + N64) * 4, stream);
  gat_edge_max<<<(int)((E8 + BT - 1) / BT), BT, 0, stream>>>(src, dst, esrc1, edst1, e1, mK1, E, 8);
  gat_edge_sum<<<(int)((E8 + BT - 1) / BT), BT, 0, stream>>>(dst, e1, mK1, s1, E, 8);
  gat_edge_msg1<<<(int)((E8 + BT - 1) / BT), BT, 0, stream>>>(src, dst, e1, s1, h1, agg1, E);
  gat_bias_relu<<<(int)((N64 + BT - 1) / BT), BT, 0, stream>>>(agg1, b1, N64);

  // ===================== Layer 2 =====================
  gat_gemm2_wmma<<<rowBlocks, 96, 0, stream>>>(agg1, W2, h2, N);
  gat_attn2<<<(N + BT - 1) / BT, BT, 0, stream>>>(h2, aS2, aD2, esrc2, edst2, N);
  hipMemsetAsync(s2, 0, (size_t)(N + N + N40) * 4, stream);
  gat_edge_max<<<(E + BT - 1) / BT, BT, 0, stream>>>(src, dst, esrc2, edst2, e2, mK2, E, 1);
  gat_edge_sum<<<(E + BT - 1) / BT, BT, 0, stream>>>(dst, e2, mK2, s2, E, 1);
  gat_edge_msg2<<<(E + BT - 1) / BT, BT, 0, stream>>>(src, dst, e2, s2, h2, agg2, E);
  gat_logsoftmax<<<(N + BT - 1) / BT, BT, 0, stream>>>(agg2, b2, out, N);
}